// JaggedLoss_19963007992160
// MI455X (gfx1250) — compile-verified
//
#include <hip/hip_runtime.h>
#include <hip/hip_bf16.h>
#include <math.h>

// ---------------------------------------------------------------------------
// JaggedLoss: masked BCE + balanced-accuracy reduction over [4096, 8192] f32.
// Bandwidth-bound (~134 MB live traffic -> ~6 us floor @ 23.3 TB/s).
// CDNA5 path used: async global->LDS copies (ASYNCcnt pipeline), wave32
// shuffle reductions, b128 vector accesses. No matrix structure -> no WMMA.
// ---------------------------------------------------------------------------

#define B_ROWS   4096
#define L_COLS   8192
#define CHUNK    1024     // elements per double-buffered LDS chunk
#define NTHREADS 256      // 8 waves (wave32)

// ---- CDNA5 async global->LDS support (probe-confirmed: builtin exists,
//      4 args, param0 = v4i __device__* (AS1), param1 = v4i __shared__* (AS3))
#if defined(__HIP_DEVICE_COMPILE__) && defined(__gfx1250__) && \
    __has_builtin(__builtin_amdgcn_global_load_async_to_lds_b128)
#define HAVE_ASYNC_LDS 1
#else
#define HAVE_ASYNC_LDS 0
#endif

#if defined(__HIP_DEVICE_COMPILE__) && defined(__gfx1250__) && !HAVE_ASYNC_LDS
#warning "gfx1250: __builtin_amdgcn_global_load_async_to_lds_b128 NOT available; using synchronous fallback"
#endif

typedef int v4i_t __attribute__((vector_size(16)));
typedef __attribute__((address_space(1))) v4i_t as1_v4i;
typedef __attribute__((address_space(3))) v4i_t as3_v4i;

__device__ __forceinline__ void async_copy_b128(const float* gsrc, float* ldst) {
#if HAVE_ASYNC_LDS
  // (global src, LDS dst, imm offset, imm cpol)
  __builtin_amdgcn_global_load_async_to_lds_b128(
      (as1_v4i*)gsrc, (as3_v4i*)ldst, 0, 0);
#else
  *(float4*)ldst = *(const float4*)gsrc;
#endif
}

#if defined(__HIP_DEVICE_COMPILE__) && defined(__gfx1250__) && \
    __has_builtin(__builtin_amdgcn_s_wait_asynccnt)
#define WAIT_ASYNC(n) __builtin_amdgcn_s_wait_asynccnt(n)
#elif HAVE_ASYNC_LDS
#define WAIT_ASYNC(n) asm volatile("s_wait_asynccnt %0" ::"i"(n) : "memory")
#else
#define WAIT_ASYNC(n) do { } while (0)
#endif

// ---------------------------------------------------------------------------
// Kernel 1: one block per row. Double-buffered async LDS pipeline; per-row
// partial loss + packed counts {tp, tn, gp, nvalid} written to workspace.
// ---------------------------------------------------------------------------
__global__ __launch_bounds__(NTHREADS)
void jagged_bce_rows(const float* __restrict__ pred,
                     const float* __restrict__ truth,
                     const int* __restrict__ lengths,
                     float* __restrict__ row_loss,
                     uint4* __restrict__ row_counts) {
  __shared__ float sp[2][CHUNK];
  __shared__ float st[2][CHUNK];
  __shared__ float wf[NTHREADS / 32];
  __shared__ uint4 wu[NTHREADS / 32];

  const int row = blockIdx.x;
  const int tid = threadIdx.x;
  const int len = lengths[row];

  if (len <= 0) {  // empty sequence contributes zeros; still must init its slot
    if (tid == 0) {
      row_loss[row]   = 0.0f;
      row_counts[row] = make_uint4(0u, 0u, 0u, 0u);
    }
    return;
  }

  const float* prow = pred  + (size_t)row * L_COLS;
  const float* trow = truth + (size_t)row * L_COLS;
  const int nchunks = (len + CHUNK - 1) / CHUNK;

  // Issue one chunk: every wave issues exactly 2 async b128 ops with full EXEC
  // (out-of-range lanes are clamped to an aligned in-row address so the
  //  per-wave ASYNCcnt bookkeeping below stays exact; garbage is masked later).
  auto issue = [&](int c, int buf) {
    const int idx = c * CHUNK + tid * 4;            // 4-aligned
    const int src = (idx < len) ? idx : 0;          // aligned, always in-row
    async_copy_b128(prow + src, &sp[buf][tid * 4]);
    async_copy_b128(trow + src, &st[buf][tid * 4]);
  };

  issue(0, 0);

  float    bce_sum = 0.0f;
  unsigned tp = 0, tn = 0, gp = 0, nv = 0;

  for (int c = 0; c < nchunks; ++c) {
    // Prefetch next chunk into the other buffer (last iter: harmless re-issue).
    const int nxt = (c + 1 < nchunks) ? (c + 1) : (nchunks - 1);
    issue(nxt, (c + 1) & 1);
    // 4 ops outstanding; <=2 means chunk c's two ops completed (in-order).
    WAIT_ASYNC(2);
    __syncthreads();

    const int   buf = c & 1;
    const int   idx = c * CHUNK + tid * 4;
    const float4 p4 = *(const float4*)&sp[buf][tid * 4];
    const float4 t4 = *(const float4*)&st[buf][tid * 4];
    const float pv[4] = {p4.x, p4.y, p4.z, p4.w};
    const float tv[4] = {t4.x, t4.y, t4.z, t4.w};

#pragma unroll
    for (int j = 0; j < 4; ++j) {
      const bool valid = (idx + j) < len;
      float p = pv[j];
      const unsigned bits = __float_as_uint(p);
      const bool finite = (bits & 0x7f800000u) != 0x7f800000u;
      p = finite ? p : 0.0f;                        // NaN/Inf -> 0 (torch-style)
      const bool  tb = tv[j] > 0.0f;                // truth in {0,1}
      const float x  = tb ? p : (1.0f - p);         // only ONE log per element
      const float lg = fmaxf(__logf(x), -100.0f);   // torch log clamp at -100
      bce_sum += valid ? -lg : 0.0f;
      const bool pb = p > 0.5f;                     // DECISIVE_CONFIDENCE
      nv += valid;
      gp += (valid && tb);
      tp += (valid && pb && tb);
      tn += (valid && !pb && !tb);
    }
    __syncthreads();  // all waves done reading buf before it is overwritten
  }
  WAIT_ASYNC(0);      // drain trailing dummy prefetch

  // wave32 down-shuffle reduction
#pragma unroll
  for (int off = 16; off > 0; off >>= 1) {
    bce_sum += __shfl_down(bce_sum, off, 32);
    tp      += __shfl_down(tp, off, 32);
    tn      += __shfl_down(tn, off, 32);
    gp      += __shfl_down(gp, off, 32);
    nv      += __shfl_down(nv, off, 32);
  }
  const int wave = tid >> 5, lane = tid & 31;
  if (lane == 0) {
    wf[wave] = bce_sum;
    wu[wave] = make_uint4(tp, tn, gp, nv);
  }
  __syncthreads();
  if (tid == 0) {
    float s = 0.0f;
    unsigned a = 0, b = 0, g = 0, n = 0;
#pragma unroll
    for (int w = 0; w < NTHREADS / 32; ++w) {
      s += wf[w];
      a += wu[w].x; b += wu[w].y; g += wu[w].z; n += wu[w].w;
    }
    row_loss[row]   = s / (float)len;   // per-sequence mean BCE
    row_counts[row] = make_uint4(a, b, g, n);
  }
}

// ---------------------------------------------------------------------------
// Kernel 2: deterministic single-block reduction over 4096 rows -> 2 scalars.
// ---------------------------------------------------------------------------
__global__ __launch_bounds__(256)
void jagged_bce_final(const float* __restrict__ row_loss,
                      const uint4* __restrict__ row_counts,
                      float* __restrict__ out) {
  __shared__ float    sf[256];
  __shared__ unsigned stp[256], stn[256], sgp[256], snv[256];
  const int tid = threadIdx.x;

  float s = 0.0f;
  unsigned tp = 0, tn = 0, gp = 0, nv = 0;
  for (int r = tid; r < B_ROWS; r += 256) {     // fixed order -> deterministic
    s += row_loss[r];
    const uint4 c = row_counts[r];
    tp += c.x; tn += c.y; gp += c.z; nv += c.w;
  }
  sf[tid] = s; stp[tid] = tp; stn[tid] = tn; sgp[tid] = gp; snv[tid] = nv;
  __syncthreads();
#pragma unroll
  for (int off = 128; off > 0; off >>= 1) {
    if (tid < off) {
      sf[tid]  += sf[tid + off];
      stp[tid] += stp[tid + off];
      stn[tid] += stn[tid + off];
      sgp[tid] += sgp[tid + off];
      snv[tid] += snv[tid + off];
    }
    __syncthreads();
  }
  if (tid == 0) {
    const float loss = sf[0] / (float)B_ROWS;
    const unsigned TP = stp[0], TN = stn[0];
    const unsigned gpos  = (sgp[0] > 1u) ? sgp[0] : 1u;
    const unsigned gnegr = snv[0] - sgp[0];
    const unsigned gneg  = (gnegr > 1u) ? gnegr : 1u;
    const float accuracy = ((float)TP / (float)gpos) * ((float)TN / (float)gneg);
    out[0] = loss;
    out[1] = accuracy;
  }
}

// ---------------------------------------------------------------------------
extern "C" void kernel_launch(void* const* d_in, const int* in_sizes, int n_in,
                              void* d_out, int out_size, void* d_ws, size_t ws_size,
                              hipStream_t stream) {
  const float* pred    = (const float*)d_in[0];
  const float* truth   = (const float*)d_in[1];
  const int*   lengths = (const int*)d_in[2];
  float*       out     = (float*)d_out;

  // Workspace layout (80 KB): [0, 4096) f32 row losses, then 4096 x uint4.
  float* row_loss   = (float*)d_ws;
  uint4* row_counts = (uint4*)((char*)d_ws + B_ROWS * sizeof(float));

  jagged_bce_rows<<<B_ROWS, NTHREADS, 0, stream>>>(pred, truth, lengths,
                                                   row_loss, row_counts);
  jagged_bce_final<<<1, 256, 0, stream>>>(row_loss, row_counts, out);
}